// ScaledDotProdAtt_29472065585554
// MI455X (gfx1250) — compile-verified
//
#include <hip/hip_runtime.h>

// ---------------------------------------------------------------------------
// Fused attention for B=2,H=16,S=2048,D=64 (fp32) with post-softmax bias.
// Outputs: values [B,H,S,D] followed by attn [B,H,S,S] (flat, fp32).
//
// MI455X reasoning:
//  - attn (512MB) must be materialized -> HBM floor ~26us @ 23.3TB/s. It is
//    write-once: emit with non-temporal (TH=NT) b128 stores so the 192MB L2
//    keeps K/V (32MB total) + peM (16MB) resident across the 128 q-tiles per
//    (b,h) that re-read them.
//  - 34.4 GFLOP fp32 via V_WMMA_F32_16X16X4_F32 (full fp32, matches the
//    jnp.float32 reference; f16/bf16 WMMA would risk tolerance).
//  - One workgroup = 16 queries, 256 threads = 8 wave32s, LDS row stride 2052
//    floats (2052 % 64 == 4) => conflict-free fragment access for all 32 lanes.
// ---------------------------------------------------------------------------

typedef __attribute__((ext_vector_type(2))) float v2f;
typedef __attribute__((ext_vector_type(4))) float v4f;
typedef __attribute__((ext_vector_type(8))) float v8f;

#define SEQ   2048
#define DH    64
#define QT    16              // queries per workgroup
#define NWAVE 8
#define KPW   (SEQ / NWAVE)   // 256 keys per wave
#define LP    2052            // padded LDS row stride for P (floats)
#define VOFF  ((size_t)2 * 16 * 2048 * 64)   // B*H*S*D, start of attn in d_out

__launch_bounds__(256, 1)
__global__ void attn_fused_kernel(const float* __restrict__ Q,
                                  const float* __restrict__ K,
                                  const float* __restrict__ V,
                                  const float* __restrict__ PE,
                                  float* __restrict__ OUT) {
  __shared__ float p[QT * LP];     // 16 x 2048 logits/probs (padded) ~131.3 KB
  __shared__ float vacc[QT * DH];  // cross-wave value accumulator      4 KB
  __shared__ float inv16[QT];

  const int tid  = threadIdx.x;
  const int wave = tid >> 5;
  const int lane = tid & 31;
  const int kh   = lane >> 4;   // 0: lanes 0-15, 1: lanes 16-31
  const int l16  = lane & 15;

  const int qt = blockIdx.x & (SEQ / QT - 1);   // 128 q-tiles per (b,h)
  const int bh = blockIdx.x >> 7;
  const int q0 = qt * QT;

  const size_t base = (size_t)bh * SEQ * DH;
  const float* qp = Q + base + (size_t)q0 * DH;
  const float* kp = K + base;
  const float* vp = V + base;

  // ---- preload Q A-fragments for all 16 K-steps (scaled by 1/sqrt(64)) ----
  // A 16x4 layout: lanes 0-15 hold {K=0,K=1} of row M=lane; lanes 16-31 {K=2,K=3}.
  v2f qa[16];
  {
    const float scale = 0.125f;
    const float* qrow = qp + l16 * DH + (kh << 1);
#pragma unroll
    for (int j = 0; j < 16; ++j) {
      v2f t = *(const v2f*)(qrow + j * 4);
      qa[j].x = t.x * scale;
      qa[j].y = t.y * scale;
    }
  }

  // ---- QK^T: each wave produces 16 key-tiles of 16 keys -------------------
  const int kbase = wave * KPW;
  for (int t = 0; t < KPW / 16; ++t) {
    const int key0 = kbase + t * 16;
    v8f acc = {};
    // B 4x16 layout: vgpr0 = row (2*kh), vgpr1 = row (2*kh+1), col = l16.
    // B[d][key] = K[key][d]  => float2 load from K row (key0+l16).
    const float* krow = kp + (size_t)(key0 + l16) * DH + (kh << 1);
#pragma unroll
    for (int j = 0; j < 16; ++j) {
      v2f b = *(const v2f*)(krow + j * 4);
      acc = __builtin_amdgcn_wmma_f32_16x16x4_f32(
          false, qa[j], false, b, (short)0, acc, false, false);
    }
    // C 16x16 layout: vgpr r = row (r + 8*kh), col = l16.
#pragma unroll
    for (int r = 0; r < 8; ++r) {
      p[(r + (kh << 3)) * LP + key0 + l16] = acc[r];
    }
  }
  __syncthreads();

  // ---- row softmax: 16 threads per row, float4 LDS sweeps -----------------
  // Address pattern row*LP + tx*4 + j*64: 16 threads of a row hit banks
  // 4*(r+tx)+{0..3} -> distinct; vectorized b128 LDS traffic.
  {
    const int tx  = tid & 15;
    const int row = tid >> 4;
    float* prow = &p[row * LP + tx * 4];
    float mx = -__builtin_inff();
#pragma unroll 4
    for (int j = 0; j < SEQ / 64; ++j) {
      v4f t = *(const v4f*)(prow + j * 64);
      mx = fmaxf(mx, fmaxf(fmaxf(t.x, t.y), fmaxf(t.z, t.w)));
    }
#pragma unroll
    for (int msk = 8; msk >= 1; msk >>= 1) mx = fmaxf(mx, __shfl_xor(mx, msk, 32));
    float s = 0.0f;
#pragma unroll 4
    for (int j = 0; j < SEQ / 64; ++j) {
      v4f t = *(const v4f*)(prow + j * 64);
      v4f e;
      e.x = __expf(t.x - mx);
      e.y = __expf(t.y - mx);
      e.z = __expf(t.z - mx);
      e.w = __expf(t.w - mx);
      *(v4f*)(prow + j * 64) = e;
      s += (e.x + e.y) + (e.z + e.w);
    }
#pragma unroll
    for (int msk = 8; msk >= 1; msk >>= 1) s += __shfl_xor(s, msk, 32);
    if (tx == 0) inv16[row] = 1.0f / s;
  }
  __syncthreads();

  // ---- normalize, emit attn (NT b128 stores), P' = attn + peM in LDS ------
  {
    const size_t attn_base = VOFF + ((size_t)bh * SEQ + q0) * SEQ;
    const float* pe = PE + (size_t)q0 * SEQ;   // peM broadcast over (b,h)
    for (int i = tid; i < QT * SEQ / 4; i += 256) {
      const int r = i >> 9;                    // (4i) / SEQ
      const int c = (i << 2) & (SEQ - 1);
      v4f pv = *(const v4f*)&p[r * LP + c];
      const float inv = inv16[r];
      pv.x *= inv; pv.y *= inv; pv.z *= inv; pv.w *= inv;
      // attn is a 512MB write-once stream: bypass cache retention (TH=NT).
      __builtin_nontemporal_store(pv, (v4f*)&OUT[attn_base + ((size_t)i << 2)]);
      v4f pe4 = *(const v4f*)&pe[(size_t)i << 2];
      pv.x += pe4.x; pv.y += pe4.y; pv.z += pe4.z; pv.w += pe4.w;
      *(v4f*)&p[r * LP + c] = pv;
    }
    for (int i = tid; i < QT * DH; i += 256) vacc[i] = 0.0f;
  }
  __syncthreads();

  // ---- PV: values = P' @ V ; wave-partial over 256 keys, 4 N-tiles --------
  {
    v8f a0 = {}, a1 = {}, a2 = {}, a3 = {};
    const int kb = wave * KPW;
    for (int ks = 0; ks < KPW / 4; ++ks) {
      const int k0 = kb + ks * 4;
      // A frag from LDS P': lanes 0-15 {k0,k0+1}, lanes 16-31 {k0+2,k0+3}.
      v2f a = *(const v2f*)&p[l16 * LP + k0 + (kh << 1)];
      // B frag from V: vgpr0 = V[k0+2*kh][n], vgpr1 = V[k0+2*kh+1][n].
      const float* vrow = vp + (size_t)(k0 + (kh << 1)) * DH + l16;
      v2f b0, b1, b2, b3;
      b0.x = vrow[0];       b0.y = vrow[DH + 0];
      b1.x = vrow[16];      b1.y = vrow[DH + 16];
      b2.x = vrow[32];      b2.y = vrow[DH + 32];
      b3.x = vrow[48];      b3.y = vrow[DH + 48];
      a0 = __builtin_amdgcn_wmma_f32_16x16x4_f32(false, a, false, b0, (short)0, a0, false, false);
      a1 = __builtin_amdgcn_wmma_f32_16x16x4_f32(false, a, false, b1, (short)0, a1, false, false);
      a2 = __builtin_amdgcn_wmma_f32_16x16x4_f32(false, a, false, b2, (short)0, a2, false, false);
      a3 = __builtin_amdgcn_wmma_f32_16x16x4_f32(false, a, false, b3, (short)0, a3, false, false);
    }
    // Reduce 8 wave-partials via LDS float atomics (ds_add_f32).
#pragma unroll
    for (int r = 0; r < 8; ++r) {
      const int m = r + (kh << 3);
      atomicAdd(&vacc[m * DH +  0 + l16], a0[r]);
      atomicAdd(&vacc[m * DH + 16 + l16], a1[r]);
      atomicAdd(&vacc[m * DH + 32 + l16], a2[r]);
      atomicAdd(&vacc[m * DH + 48 + l16], a3[r]);
    }
  }
  __syncthreads();

  // ---- contiguous 1024-float (256 x b128, NT) store of the values tile ----
  {
    const size_t vbase = ((size_t)bh * SEQ + q0) * DH;
    v4f t = *(const v4f*)&vacc[tid * 4];
    __builtin_nontemporal_store(t, (v4f*)&OUT[vbase + (size_t)tid * 4]);
  }
}

extern "C" void kernel_launch(void* const* d_in, const int* in_sizes, int n_in,
                              void* d_out, int out_size, void* d_ws, size_t ws_size,
                              hipStream_t stream) {
  const float* q  = (const float*)d_in[0];
  const float* k  = (const float*)d_in[1];
  const float* v  = (const float*)d_in[2];
  const float* pe = (const float*)d_in[3];
  float* out = (float*)d_out;

  // Grid: B*H * (S/QT) = 32 * 128 = 4096 workgroups, 256 threads (8 wave32s).
  attn_fused_kernel<<<dim3(4096), dim3(256), 0, stream>>>(q, k, v, pe, out);
}